// Resizer_79405355369073
// MI455X (gfx1250) — compile-verified
//
#include <hip/hip_runtime.h>

typedef __attribute__((ext_vector_type(2))) float v2f;
typedef __attribute__((ext_vector_type(8))) float v8f;

#define TILE 16
#define MAXSPAN 64
#define APAD (MAXSPAN + 1)   // pad row stride to kill 16-way bank conflicts

// ---------------------------------------------------------------------------
// Pass 1: resize along H.  out[b][ho][w] = sum_hi Wh[ho][hi] * x[b][hi][w]
// Tile = 16 output rows (ho) x 16 contiguous columns (w).  A = dense weights,
// B = input tile.  D = A x B via V_WMMA_F32_16X16X4_F32.
// ---------------------------------------------------------------------------
__global__ __launch_bounds__(32)
void resize_h_wmma(const float* __restrict__ x,    // [B][H_in][W]
                   const float* __restrict__ w0,   // [K][HO]
                   const int*   __restrict__ fov0, // [K][HO]
                   float* __restrict__ out,        // [B][HO][W]
                   int B, int H_in, int W, int HO, int K)
{
    __shared__ float Wh[TILE][APAD];     // A: [ho_local][hi - rmin]
    __shared__ float Xs[MAXSPAN][TILE];  // B: [hi - rmin][w_local]
    __shared__ int s_min, s_max;

    const int wTiles = W / TILE;
    const int hTiles = HO / TILE;
    int tid = blockIdx.x;
    int b   = tid / (wTiles * hTiles);
    int rem = tid % (wTiles * hTiles);
    int ho_base = (rem / wTiles) * TILE;
    int w_base  = (rem % wTiles) * TILE;
    int lane = threadIdx.x;

    if (lane == 0) { s_min = 0x7fffffff; s_max = -0x7fffffff; }
    __syncthreads();
    for (int i = lane; i < K * TILE; i += 32) {
        int r = fov0[(i / TILE) * HO + ho_base + (i % TILE)];
        atomicMin(&s_min, r);
        atomicMax(&s_max, r);
    }
    __syncthreads();
    const int rmin = s_min;
    int span = s_max - rmin + 1;
    if (span > MAXSPAN) span = MAXSPAN;

    // zero LDS (incl. pad; guarantees 0*garbage never becomes NaN)
    for (int i = lane; i < TILE * APAD; i += 32)       ((float*)Wh)[i] = 0.0f;
    for (int i = lane; i < MAXSPAN * TILE; i += 32)    ((float*)Xs)[i] = 0.0f;
    __syncthreads();

    // scatter taps into dense weight matrix (mirror boundary can collide -> atomic)
    for (int i = lane; i < K * TILE; i += 32) {
        int k = i / TILE, j = i % TILE;
        int ho = ho_base + j;
        int r  = fov0[k * HO + ho] - rmin;
        if (r >= 0 && r < MAXSPAN)
            atomicAdd(&Wh[j][r], w0[k * HO + ho]);
    }

    // stage input tile: rows [rmin, rmin+span) x 16 cols, coalesced 64B per half-wave
    const float* xb = x + (size_t)b * H_in * W;
    for (int i = lane; i < span * TILE; i += 32) {
        int r = i / TILE, c = i % TILE;
        Xs[r][c] = xb[(size_t)(rmin + r) * W + (w_base + c)];
    }
    __syncthreads();

    const int m  = lane & 15;   // M for A / N for B,D
    const int kh = lane >> 4;   // which K-pair this half-wave holds
    v8f acc = {};
    int steps = (span + 3) >> 2;
    for (int kk = 0; kk < steps; ++kk) {
        int kidx = kk * 4 + 2 * kh;
        v2f a, bb;
        a.x  = Wh[m][kidx];     a.y  = Wh[m][kidx + 1];
        bb.x = Xs[kidx][m];     bb.y = Xs[kidx + 1][m];
        acc = __builtin_amdgcn_wmma_f32_16x16x4_f32(
                  false, a, false, bb, (short)0, acc, false, false);
    }

    // D layout: lane holds column n=m; VGPR r holds row r + 8*kh
    float* ob = out + (size_t)b * HO * W + (size_t)ho_base * W + w_base;
#pragma unroll
    for (int r = 0; r < 8; ++r)
        ob[(size_t)(r + 8 * kh) * W + m] = acc[r];
}

// ---------------------------------------------------------------------------
// Pass 2: resize along W.  out[r][co] = sum_ci t[r][ci] * Wd[ci][co]
// Tile = 16 rows x 16 output cols.  A = input tile, B = dense weights.
// ---------------------------------------------------------------------------
__global__ __launch_bounds__(32)
void resize_w_wmma(const float* __restrict__ t,    // [R][W_in]
                   const float* __restrict__ w1,   // [K][WO]
                   const int*   __restrict__ fov1, // [K][WO]
                   float* __restrict__ out,        // [R][WO]
                   int R, int W_in, int WO, int K)
{
    __shared__ float As[TILE][APAD];     // A: [row_local][ci - cmin]
    __shared__ float Wd[MAXSPAN][TILE];  // B: [ci - cmin][co_local]
    __shared__ int s_min, s_max;

    const int cTiles = WO / TILE;
    int tid = blockIdx.x;
    int r_base  = (tid / cTiles) * TILE;
    int co_base = (tid % cTiles) * TILE;
    int lane = threadIdx.x;

    if (lane == 0) { s_min = 0x7fffffff; s_max = -0x7fffffff; }
    __syncthreads();
    for (int i = lane; i < K * TILE; i += 32) {
        int c = fov1[(i / TILE) * WO + co_base + (i % TILE)];
        atomicMin(&s_min, c);
        atomicMax(&s_max, c);
    }
    __syncthreads();
    const int cmin = s_min;
    int span = s_max - cmin + 1;
    if (span > MAXSPAN) span = MAXSPAN;

    for (int i = lane; i < TILE * APAD; i += 32)       ((float*)As)[i] = 0.0f;
    for (int i = lane; i < MAXSPAN * TILE; i += 32)    ((float*)Wd)[i] = 0.0f;
    __syncthreads();

    for (int i = lane; i < K * TILE; i += 32) {
        int k = i / TILE, j = i % TILE;
        int co = co_base + j;
        int c  = fov1[k * WO + co] - cmin;
        if (c >= 0 && c < MAXSPAN)
            atomicAdd(&Wd[c][j], w1[k * WO + co]);
    }

    // stage input tile rows: fully coalesced along ci
    for (int r = 0; r < TILE; ++r) {
        const float* row = t + (size_t)(r_base + r) * W_in + cmin;
        for (int c = lane; c < span; c += 32)
            As[r][c] = row[c];
    }
    __syncthreads();

    const int m  = lane & 15;
    const int kh = lane >> 4;
    v8f acc = {};
    int steps = (span + 3) >> 2;
    for (int kk = 0; kk < steps; ++kk) {
        int kidx = kk * 4 + 2 * kh;
        v2f a, bb;
        a.x  = As[m][kidx];     a.y  = As[m][kidx + 1];
        bb.x = Wd[kidx][m];     bb.y = Wd[kidx + 1][m];
        acc = __builtin_amdgcn_wmma_f32_16x16x4_f32(
                  false, a, false, bb, (short)0, acc, false, false);
    }

    float* ob = out + (size_t)r_base * WO + co_base;
#pragma unroll
    for (int r = 0; r < 8; ++r)
        ob[(size_t)(r + 8 * kh) * WO + m] = acc[r];
}

// ---------------------------------------------------------------------------
extern "C" void kernel_launch(void* const* d_in, const int* in_sizes, int n_in,
                              void* d_out, int out_size, void* d_ws, size_t ws_size,
                              hipStream_t stream) {
    const float* x    = (const float*)d_in[0];
    const float* w0   = (const float*)d_in[1];
    const int*   fov0 = (const int*)  d_in[2];
    const float* w1   = (const float*)d_in[3];
    const int*   fov1 = (const int*)  d_in[4];
    float* out = (float*)d_out;

    const int N = 8, C = 3, H = 1024, W = 1024;
    const int HO = 512, WO = 512;
    const int B  = N * C;
    const int K0 = in_sizes[1] / HO;   // taps for H pass
    const int K1 = in_sizes[3] / WO;   // taps for W pass

    float* tmp = (float*)d_ws;         // [B][HO][W] intermediate (50.3 MB)

    int g1 = B * (HO / TILE) * (W / TILE);
    resize_h_wmma<<<g1, 32, 0, stream>>>(x, w0, fov0, tmp, B, H, W, HO, K0);

    int g2 = ((B * HO) / TILE) * (WO / TILE);
    resize_w_wmma<<<g2, 32, 0, stream>>>(tmp, w1, fov1, out, B * HO, W, WO, K1);
}